// DotScoringComponent_83683142795691
// MI455X (gfx1250) — compile-verified
//
#include <hip/hip_runtime.h>
#include <cmath>

typedef __attribute__((ext_vector_type(2))) float v2f;
typedef __attribute__((ext_vector_type(8))) float v8f;

#define M_TOTAL 4096      // B*S
#define HIDDEN_ 1024
#define ENC_ 256
#define NNODES 21844
#define NTIMES 48

static __device__ __forceinline__ int imin(int a, int b) { return a < b ? a : b; }

static __device__ __forceinline__ v8f wmma_f32(v2f a, v2f b, v8f c) {
    // V_WMMA_F32_16X16X4_F32 : D = A(16x4) x B(4x16) + C(16x16)
    return __builtin_amdgcn_wmma_f32_16x16x4_f32(
        /*neg_a=*/false, a, /*neg_b=*/false, b,
        /*c_mod=*/(short)0, c, /*reuse_a=*/false, /*reuse_b=*/false);
}

// ---------------------------------------------------------------------------
// P2[4096,256] = prefix[4096,1024] @ W_key[1024,256]
// One wave per 16(M) x 64(N) tile; 4 accumulators reuse the A fragment.
// ---------------------------------------------------------------------------
__global__ void gemm_p2(const float* __restrict__ A, const float* __restrict__ Bw,
                        float* __restrict__ P2) {
    const int l = threadIdx.x;
    const int lo = l & 15, half = l >> 4;
    const int m0 = blockIdx.y * 16;
    const int n0 = blockIdx.x * 64;

    v8f c0 = {}, c1 = {}, c2 = {}, c3 = {};
    const float* arow = A + (size_t)(m0 + lo) * HIDDEN_;

#pragma unroll 2
    for (int k0 = 0; k0 < HIDDEN_; k0 += 4) {
        const int k = k0 + 2 * half;                    // this lane's K pair
        v2f a = *(const v2f*)(arow + k);                // A[m=lo][k], A[m][k+1]
        const float* bp = Bw + (size_t)k * ENC_ + n0 + lo;
        v2f b0 = { bp[0],  bp[ENC_] };                  // B[k][n], B[k+1][n]
        v2f b1 = { bp[16], bp[16 + ENC_] };
        v2f b2 = { bp[32], bp[32 + ENC_] };
        v2f b3 = { bp[48], bp[48 + ENC_] };
        c0 = wmma_f32(a, b0, c0);
        c1 = wmma_f32(a, b1, c1);
        c2 = wmma_f32(a, b2, c2);
        c3 = wmma_f32(a, b3, c3);
    }

#pragma unroll
    for (int v = 0; v < 8; ++v) {
        const int row = m0 + v + 8 * half;
        float* op = P2 + (size_t)row * ENC_ + n0 + lo;
        op[0] = c0[v]; op[16] = c1[v]; op[32] = c2[v]; op[48] = c3[v];
    }
}

// ---------------------------------------------------------------------------
// r[row] = dot(prefix[row,:], b_key)  (per-row scalar shared across all nodes)
// ---------------------------------------------------------------------------
__global__ void rowdot(const float* __restrict__ A, const float* __restrict__ bk,
                       float* __restrict__ r) {
    const int wave = threadIdx.x >> 5, lane = threadIdx.x & 31;
    const int row = blockIdx.x * 8 + wave;
    const float* p = A + (size_t)row * HIDDEN_;
    float s = 0.f;
    for (int k = lane; k < HIDDEN_; k += 32) s += p[k] * bk[k];
    for (int o = 16; o > 0; o >>= 1) s += __shfl_xor(s, o);
    if (lane == 0) r[row] = s;
}

// ---------------------------------------------------------------------------
// scores block: out[4096, ncols] = P2[4096,256] @ emb[col_off:.., 256]^T + r
// B = emb^T, so B fragments are contiguous float2 along emb rows.
// ---------------------------------------------------------------------------
__global__ void gemm_scores(const float* __restrict__ P2, const float* __restrict__ emb,
                            const float* __restrict__ r, float* __restrict__ out,
                            int col_off, int ncols) {
    const int l = threadIdx.x;
    const int lo = l & 15, half = l >> 4;
    const int m0 = blockIdx.y * 16;
    const int n0 = blockIdx.x * 64;

    v8f c0 = {}, c1 = {}, c2 = {}, c3 = {};
    const float* arow = P2 + (size_t)(m0 + lo) * ENC_;

    const int ng = col_off + n0 + lo;                   // global node index, lane column
    const float* e0 = emb + (size_t)imin(ng,      NNODES - 1) * ENC_;
    const float* e1 = emb + (size_t)imin(ng + 16, NNODES - 1) * ENC_;
    const float* e2 = emb + (size_t)imin(ng + 32, NNODES - 1) * ENC_;
    const float* e3 = emb + (size_t)imin(ng + 48, NNODES - 1) * ENC_;

#pragma unroll 4
    for (int k0 = 0; k0 < ENC_; k0 += 4) {
        const int k = k0 + 2 * half;
        v2f a  = *(const v2f*)(arow + k);
        v2f b0 = *(const v2f*)(e0 + k);
        v2f b1 = *(const v2f*)(e1 + k);
        v2f b2 = *(const v2f*)(e2 + k);
        v2f b3 = *(const v2f*)(e3 + k);
        c0 = wmma_f32(a, b0, c0);
        c1 = wmma_f32(a, b1, c1);
        c2 = wmma_f32(a, b2, c2);
        c3 = wmma_f32(a, b3, c3);
    }

#pragma unroll
    for (int v = 0; v < 8; ++v) {
        const int row = m0 + v + 8 * half;
        const float rv = r[row];
        float* op = out + (size_t)row * ncols;
        const int nl = n0 + lo;
        if (nl      < ncols) op[nl]      = c0[v] + rv;
        if (nl + 16 < ncols) op[nl + 16] = c1[v] + rv;
        if (nl + 32 < ncols) op[nl + 32] = c2[v] + rv;
        if (nl + 48 < ncols) op[nl + 48] = c3[v] + rv;
    }
}

// ---------------------------------------------------------------------------
// In-place log_softmax over each row of data[4096, ncols]. One block per row.
// ---------------------------------------------------------------------------
__global__ void logsoftmax_rows(float* __restrict__ data, int ncols) {
    __shared__ float sred[8];
    const size_t base = (size_t)blockIdx.x * (size_t)ncols;
    const int tid = threadIdx.x, lane = tid & 31, wv = tid >> 5;

    float m = -3.4e38f;
    for (int i = tid; i < ncols; i += 256) m = fmaxf(m, data[base + i]);
    for (int o = 16; o > 0; o >>= 1) m = fmaxf(m, __shfl_xor(m, o));
    if (lane == 0) sred[wv] = m;
    __syncthreads();
    const float mall = fmaxf(fmaxf(fmaxf(sred[0], sred[1]), fmaxf(sred[2], sred[3])),
                             fmaxf(fmaxf(sred[4], sred[5]), fmaxf(sred[6], sred[7])));
    __syncthreads();

    float s = 0.f;
    for (int i = tid; i < ncols; i += 256) s += __expf(data[base + i] - mall);
    for (int o = 16; o > 0; o >>= 1) s += __shfl_xor(s, o);
    if (lane == 0) sred[wv] = s;
    __syncthreads();
    const float stot = sred[0] + sred[1] + sred[2] + sred[3] +
                       sred[4] + sred[5] + sred[6] + sred[7];
    const float lse = mall + __logf(stot);
    for (int i = tid; i < ncols; i += 256) data[base + i] -= lse;
}

// ---------------------------------------------------------------------------
// time head: log_softmax(prefix @ W_time^T + b_time)  -> out[4096,48]
// One 64-thread block per row; prefix row staged in LDS.
// ---------------------------------------------------------------------------
__global__ void time_kernel(const float* __restrict__ prefix, const float* __restrict__ Wt,
                            const float* __restrict__ bt, float* __restrict__ out) {
    __shared__ float sp[HIDDEN_];
    __shared__ float logits[NTIMES];
    __shared__ float slse;
    const int row = blockIdx.x, t = threadIdx.x;

    for (int k = t; k < HIDDEN_; k += 64) sp[k] = prefix[(size_t)row * HIDDEN_ + k];
    __syncthreads();

    if (t < NTIMES) {
        float acc = bt[t];
        const float* w = Wt + (size_t)t * HIDDEN_;
        for (int k = 0; k < HIDDEN_; ++k) acc += sp[k] * w[k];
        logits[t] = acc;
    }
    __syncthreads();
    if (t == 0) {
        float m = -3.4e38f;
        for (int i = 0; i < NTIMES; ++i) m = fmaxf(m, logits[i]);
        float s = 0.f;
        for (int i = 0; i < NTIMES; ++i) s += __expf(logits[i] - m);
        slse = m + __logf(s);
    }
    __syncthreads();
    if (t < NTIMES) out[(size_t)row * NTIMES + t] = logits[t] - slse;
}

// ---------------------------------------------------------------------------
extern "C" void kernel_launch(void* const* d_in, const int* in_sizes, int n_in,
                              void* d_out, int out_size, void* d_ws, size_t ws_size,
                              hipStream_t stream) {
    const float* prefix = (const float*)d_in[0];   // [4096,1024]
    const float* emb    = (const float*)d_in[1];   // [21844,256]
    const float* Wk     = (const float*)d_in[2];   // [1024,256]
    const float* bk     = (const float*)d_in[3];   // [1024]
    const float* Wt     = (const float*)d_in[4];   // [48,1024]
    const float* bt     = (const float*)d_in[5];   // [48]
    float* out = (float*)d_out;

    float* P2 = (float*)d_ws;                      // [4096,256]
    float* r  = P2 + (size_t)M_TOTAL * ENC_;       // [4096]

    // Stage A: P2 = prefix @ W_key ; r = prefix . b_key
    gemm_p2<<<dim3(ENC_ / 64, M_TOTAL / 16), 32, 0, stream>>>(prefix, Wk, P2);
    rowdot<<<M_TOTAL / 8, 256, 0, stream>>>(prefix, bk, r);

    // Stage B: per-depth score block -> log_softmax in place
    static const int depth_sizes[7] = {4, 16, 64, 256, 1024, 4096, 16384};
    int col_off = 0;
    size_t out_off = 0;
    for (int d = 0; d < 7; ++d) {
        const int n = depth_sizes[d];
        dim3 grid((n + 63) / 64, M_TOTAL / 16);
        gemm_scores<<<grid, 32, 0, stream>>>(P2, emb, r, out + out_off, col_off, n);
        logsoftmax_rows<<<M_TOTAL, 256, 0, stream>>>(out + out_off, n);
        col_off += n;
        out_off += (size_t)M_TOTAL * (size_t)n;
    }

    // Time head (independent of stages A/B)
    time_kernel<<<M_TOTAL, 64, 0, stream>>>(prefix, Wt, bt, out + out_off);
}